// CenterEmpiricalPriorMemory_61108794688227
// MI455X (gfx1250) — compile-verified
//
#include <hip/hip_runtime.h>

#define NUM_C   1024
#define DIM     256
#define NROWS   131072
#define MOM     0.05f

#define NSEG    8
#define SEGLEN  (NROWS / NSEG)      // 16384 labels scanned per block
#define CAP     2048                // compacted-row list capacity (expected ~256)
#define CT      16                  // centers per tile (WMMA M)
#define NWAVE   4
#define TPB     (NWAVE * 32)        // 128 threads
#define SENTINEL 16                 // decodes to row 0, m=16 (never matches -> zero column)

typedef float v2f __attribute__((ext_vector_type(2)));
typedef float v8f __attribute__((ext_vector_type(8)));

__global__ void zero_ws_kernel(float* ws, int n) {
    int i = blockIdx.x * blockDim.x + threadIdx.x;
    if (i < n) ws[i] = 0.0f;
}

// retrieved[row][:] = slots[clip(labels[row])][:]   (block per row, float4 copy)
__global__ void gather_kernel(const float4* __restrict__ slots4,
                              const int* __restrict__ labels,
                              float4* __restrict__ out4) {
    int row = blockIdx.x;
    int c = labels[row];
    c = c < 0 ? 0 : (c > NUM_C - 1 ? NUM_C - 1 : c);
    out4[(size_t)row * (DIM / 4) + threadIdx.x] = slots4[(size_t)c * (DIM / 4) + threadIdx.x];
}

// Segment-sum via one-hot f32 WMMA: block (ct, seg) owns 16 centers, scans one
// label segment, compacts matching rows, and accumulates
// D[16c x 16d] += OneHot(16c x 4rows) * States(4rows x 16d) with v_wmma_f32_16x16x4_f32.
__global__ void __launch_bounds__(TPB)
segsum_kernel(const float* __restrict__ states,
              const int* __restrict__ labels,
              float* __restrict__ seg_sum,   // [NUM_C * DIM] in ws (zeroed)
              float* __restrict__ bcnt) {    // [NUM_C] in ws (zeroed)
    __shared__ int   list[CAP + 4];          // +4 sentinel pad -> branch-free reads
    __shared__ int   nmatch;
    __shared__ float tile[CT * DIM];
    __shared__ float cnt[CT];

    const int ct   = blockIdx.x;        // 0..63
    const int base = ct * CT;
    const int seg  = blockIdx.y;        // 0..NSEG-1
    const int tid  = threadIdx.x;

    if (tid == 0) nmatch = 0;
    if (tid < CT) cnt[tid] = 0.0f;
    for (int t = tid; t < CT * DIM; t += TPB) tile[t] = 0.0f;
    __syncthreads();

    // ---- Phase 1: scan labels, compact matching rows into LDS ----
    const int s0 = seg * SEGLEN;
    for (int i = tid; i < SEGLEN; i += TPB) {
        int idx = s0 + i;
        int l = labels[idx];
        l = l < 0 ? 0 : (l > NUM_C - 1 ? NUM_C - 1 : l);
        unsigned m = (unsigned)(l - base);
        if (m < CT) {
            atomicAdd(&cnt[m], 1.0f);
            int pos = atomicAdd(&nmatch, 1);
            if (pos < CAP) list[pos] = (idx << 5) | (int)m;
        }
    }
    __syncthreads();

    // Pad 4 sentinels past the end so the WMMA loop reads unconditionally.
    if (tid < 4) {
        int nmc = nmatch; if (nmc > CAP) nmc = CAP;
        list[nmc + tid] = SENTINEL;
    }
    __syncthreads();

    // Force block/wave-uniform values into SGPRs -> pure scalar loop control,
    // EXEC stays all-1s through the WMMA region (ISA requirement).
    int nm = __builtin_amdgcn_readfirstlane(nmatch);
    if (nm > CAP) nm = CAP;
    const int wv_s    = __builtin_amdgcn_readfirstlane(tid >> 5);
    const int ngroups = (nm + 3) >> 2;

    const int lane = tid & 31;
    const int half = lane >> 4;     // K-group: 0 -> rows 0,1 ; 1 -> rows 2,3
    const int n16  = lane & 15;     // M for A-matrix, N for B/D matrices

    v8f acc[16];
    #pragma unroll
    for (int d = 0; d < 16; ++d) acc[d] = (v8f){0.f,0.f,0.f,0.f,0.f,0.f,0.f,0.f};

    // ---- Phase 2: WMMA accumulate, 4 compacted rows per step ----
    for (int g = wv_s; g < ngroups; g += NWAVE) {
        int e0 = list[4*g + 0];
        int e1 = list[4*g + 1];
        int e2 = list[4*g + 2];
        int e3 = list[4*g + 3];

        int ea = half ? e2 : e0;   // K = 2*half
        int eb = half ? e3 : e1;   // K = 2*half + 1
        int ra = ea >> 5;
        int rb = eb >> 5;
        int ma = ea & 31;          // 16 (sentinel) never matches n16 in 0..15
        int mb = eb & 31;

        // A-matrix (16x4 f32, one-hot): lane n16 = row M, VGPR0/1 = K(2*half), K(2*half+1)
        v2f a;
        a.x = (ma == n16) ? 1.0f : 0.0f;
        a.y = (mb == n16) ? 1.0f : 0.0f;

        const float* pa = states + (size_t)ra * DIM;
        const float* pb = states + (size_t)rb * DIM;

        #pragma unroll
        for (int dt = 0; dt < 16; ++dt) {
            // B-matrix (4x16 f32): VGPR0 = row K(2*half), VGPR1 = row K(2*half+1), N = n16
            v2f b;
            b.x = pa[dt * 16 + n16];
            b.y = pb[dt * 16 + n16];
            acc[dt] = __builtin_amdgcn_wmma_f32_16x16x4_f32(
                false, a, false, b, (short)0, acc[dt], false, false);
        }
    }

    // ---- Phase 3: cross-wave reduce via LDS f32 atomics ----
    // C/D layout: VGPR j -> M = j + 8*half, N = n16 ; dim = dt*16 + n16
    #pragma unroll
    for (int dt = 0; dt < 16; ++dt) {
        #pragma unroll
        for (int j = 0; j < 8; ++j) {
            int m = j + 8 * half;
            atomicAdd(&tile[m * DIM + dt * 16 + n16], acc[dt][j]);
        }
    }
    __syncthreads();

    // ---- Phase 4: flush tile to L2-resident scratch with f32 atomics ----
    for (int t = tid; t < CT * DIM; t += TPB)
        unsafeAtomicAdd(&seg_sum[(size_t)base * DIM + t], tile[t]);
    if (tid < CT)
        unsafeAtomicAdd(&bcnt[base + tid], cnt[tid]);
}

__global__ void finalize_kernel(const float* __restrict__ slots,
                                const float* __restrict__ counts,
                                const float* __restrict__ seg_sum,
                                const float* __restrict__ bcnt,
                                float* __restrict__ new_slots,
                                float* __restrict__ new_counts) {
    int c = blockIdx.x;          // 0..1023
    int t = threadIdx.x;         // 0..63 (float4 lanes)
    float bc = bcnt[c];
    float cn = counts[c];
    float inv = 1.0f / fmaxf(bc, 1.0f);
    bool present = bc > 0.0f;
    bool cold = cn <= 0.0f;

    const float4* s4  = (const float4*)seg_sum + (size_t)c * (DIM / 4);
    const float4* sl4 = (const float4*)slots   + (size_t)c * (DIM / 4);
    float4 sum = s4[t];
    float4 sl  = sl4[t];

    float4 r;
    {
        float mx = sum.x * inv, my = sum.y * inv, mz = sum.z * inv, mw = sum.w * inv;
        float ex = (1.0f - MOM) * sl.x + MOM * mx;
        float ey = (1.0f - MOM) * sl.y + MOM * my;
        float ez = (1.0f - MOM) * sl.z + MOM * mz;
        float ew = (1.0f - MOM) * sl.w + MOM * mw;
        float cx = cold ? mx : ex, cy = cold ? my : ey, cz = cold ? mz : ez, cw = cold ? mw : ew;
        r.x = present ? cx : sl.x;
        r.y = present ? cy : sl.y;
        r.z = present ? cz : sl.z;
        r.w = present ? cw : sl.w;
    }
    ((float4*)new_slots)[(size_t)c * (DIM / 4) + t] = r;

    if (t == 0) new_counts[c] = cn + bc;
}

extern "C" void kernel_launch(void* const* d_in, const int* in_sizes, int n_in,
                              void* d_out, int out_size, void* d_ws, size_t ws_size,
                              hipStream_t stream) {
    const float* states = (const float*)d_in[0];
    const float* slots  = (const float*)d_in[1];
    const float* counts = (const float*)d_in[2];
    const int*   labels = (const int*)d_in[3];

    float* out        = (float*)d_out;
    float* retrieved  = out;                                   // [N, DIM]
    float* new_slots  = out + (size_t)NROWS * DIM;             // [C, DIM]
    float* new_counts = new_slots + (size_t)NUM_C * DIM;       // [C]

    float* ws_sum = (float*)d_ws;                              // [C*DIM]
    float* ws_cnt = ws_sum + (size_t)NUM_C * DIM;              // [C]

    int nz = NUM_C * DIM + NUM_C;
    zero_ws_kernel<<<(nz + 255) / 256, 256, 0, stream>>>(ws_sum, nz);

    gather_kernel<<<NROWS, DIM / 4, 0, stream>>>(
        (const float4*)slots, labels, (float4*)retrieved);

    segsum_kernel<<<dim3(NUM_C / CT, NSEG), TPB, 0, stream>>>(
        states, labels, ws_sum, ws_cnt);

    finalize_kernel<<<NUM_C, DIM / 4, 0, stream>>>(
        slots, counts, ws_sum, ws_cnt, new_slots, new_counts);
}